// Set2SetPooling_32049045963321
// MI455X (gfx1250) — compile-verified
//
#include <hip/hip_runtime.h>
#include <math.h>

typedef __attribute__((ext_vector_type(2))) float v2f;
typedef __attribute__((ext_vector_type(8))) float v8f;
typedef __attribute__((ext_vector_type(4))) unsigned int u32x4;
typedef __attribute__((ext_vector_type(4))) int i32x4;
typedef __attribute__((ext_vector_type(8))) int i32x8;

#define DDIM 512
#define NSTEPS 6
#define NB 160            // attend grid size
#define ROWS 32           // rows per TDM tile
#define LSTRIDE 516       // row stride in LDS after TDM padding (512 + 2x2 DWORDs)
#define XBUF (ROWS * LSTRIDE)

// workspace layout (float offsets)
#define H_OFF     0                     // h[512]
#define C_OFF     512                   // c[512]
#define CTX_OFF   1024                  // contexts[6*512]
#define GATES_OFF 4096                  // gates[2048]
#define PLOG_OFF  6144                  // projection logits[512]
#define PM_OFF    6656                  // per-block running max [NB]
#define PS_OFF    6912                  // per-block running sum [NB]
#define PCTX_OFF  7168                  // per-block ctx partials [NB*512]

__global__ void s2s_init_kernel(float* ws) {
    int t = threadIdx.x;
    if (t < 2 * DDIM) ws[t] = 0.0f;     // zero h and c
}

// TDM: DMA one ROWS x 512 fp32 tile (row-major, stride 512) global -> LDS,
// inserting 2 pad DWORDs every 256 DWORDs (LDS row stride becomes 516 floats).
// OOB rows (beyond rows_rem) read as zero per D# tensor_dim1 bounds.
__device__ __forceinline__ void tdm_load_tile(const float* gaddr, unsigned lds_off,
                                              unsigned rows_rem) {
    unsigned long long ga = (unsigned long long)(uintptr_t)gaddr;
    u32x4 g0;
    g0[0] = 1u;                                            // count=1, user descriptor
    g0[1] = lds_off;                                       // LDS byte address
    g0[2] = (unsigned)(ga & 0xFFFFFFFFu);                  // global_addr[31:0]
    g0[3] = (unsigned)((ga >> 32) & 0x01FFFFFFu) | (2u << 30); // addr[56:32], type=2
    i32x8 g1;
    g1[0] = (2 << 16)      // data_size = 4 bytes
          | (1 << 20)      // pad_enable
          | (7 << 22)      // pad_interval: every 256 DWORDs
          | (1 << 25);     // pad_amount: 2 DWORDs
    g1[1] = (int)((512u & 0xFFFFu) << 16);                 // tensor_dim0[15:0]=512
    g1[2] = (int)((rows_rem & 0xFFFFu) << 16);             // dim0[31:16]=0 | tensor_dim1 lo16
    g1[3] = (int)(((rows_rem >> 16) & 0xFFFFu) | (512u << 16)); // dim1 hi16 | tile_dim0=512
    g1[4] = ROWS;                                          // tile_dim1 | tile_dim2=0
    g1[5] = 512;                                           // tensor_dim0_stride[31:0]
    g1[6] = 0;                                             // stride hi | dim1_stride lo
    g1[7] = 0;
    i32x4 gz = {0, 0, 0, 0};
#if defined(__clang_major__) && (__clang_major__ >= 23)
    i32x8 gz8 = {0, 0, 0, 0, 0, 0, 0, 0};
    __builtin_amdgcn_tensor_load_to_lds(g0, g1, gz, gz, gz8, 0);
#else
    __builtin_amdgcn_tensor_load_to_lds(g0, g1, gz, gz, 0);
#endif
}

// Fused pass over x: TDM double-buffered staging, logits via WMMA f32 16x16x4,
// online softmax, rank-1 ctx accumulation. x is read from HBM exactly once per step.
__global__ void s2s_attend_kernel(const float* __restrict__ x, float* __restrict__ ws, int N) {
    extern __shared__ float smem[];
    float* xs0   = smem;                 // XBUF floats (TDM-padded tile, buffer 0)
    float* xs1   = smem + XBUF;          // XBUF floats (buffer 1)
    float* hs    = smem + 2 * XBUF;      // 512
    float* lpart = hs + DDIM;            // 4 K-quarters * 32 rows
    float* prob  = lpart + 4 * ROWS;     // 32

    const int tid  = threadIdx.x;
    const int lane = tid & 31;
    const int wave = tid >> 5;
    const int g    = wave & 1;           // row group: rows 16g..16g+15
    const int kq   = wave >> 1;          // K quarter: 128 cols
    const int kbase = 128 * kq;
    const int kadj  = (kq >= 2) ? 2 : 0; // mid-row TDM pad offset for cols >= 256

    for (int i = tid; i < DDIM; i += 256) hs[i] = ws[H_OFF + i];

    const int d0   = 2 * tid;            // this thread owns ctx dims d0, d0+1
    const int dadj = (d0 >= 256) ? 2 : 0;
    float m_run = -INFINITY, s_run = 0.0f, ctx0 = 0.0f, ctx1 = 0.0f;

    const int nChunks = (N + ROWS - 1) / ROWS;

    // prologue: kick off first tile (wave 0 only; TDM ignores EXEC so only wave 0
    // may reach the instruction)
    if (blockIdx.x < nChunks && tid < 32) {
        tdm_load_tile(x + (long)blockIdx.x * ROWS * DDIM,
                      (unsigned)(uintptr_t)xs0,
                      (unsigned)(N - (long)blockIdx.x * ROWS));
    }
    __syncthreads();   // hs visible

    int buf = 0;
    for (int chunk = blockIdx.x; chunk < nChunks; chunk += NB, buf ^= 1) {
        const long rowBase = (long)chunk * ROWS;
        float* xs = buf ? xs1 : xs0;
        // issue next tile into the other buffer, then wait for the current one.
        const int next = chunk + NB;
        if (next < nChunks) {
            if (tid < 32) {
                tdm_load_tile(x + (long)next * ROWS * DDIM,
                              (unsigned)(uintptr_t)(buf ? xs0 : xs1),
                              (unsigned)(N - (long)next * ROWS));
                __builtin_amdgcn_s_wait_tensorcnt(1);   // current tile done, next in flight
            }
        } else {
            if (tid < 32) __builtin_amdgcn_s_wait_tensorcnt(0);
        }
        // L2 warm-up for the tile after next
        {
            long pfBase = rowBase + 2L * NB * ROWS;
            if (pfBase + ROWS <= N) __builtin_prefetch(x + pfBase * DDIM + tid * 64, 0, 1);
        }
        __syncthreads();   // tile `buf` ready for all waves

        // logits: wave (g,kq) computes rows 16g..16g+15 over its 128-wide K quarter
        // A frag: lanes 0-15 hold M=lane K={0,1}; lanes 16-31 hold M=lane-16 K={2,3}
        const float* arow = xs + (16 * g + (lane & 15)) * LSTRIDE + kbase + kadj + 2 * (lane >> 4);
        const float* brow = hs + kbase + 2 * (lane >> 4);   // B[k][n] = h[k] broadcast
        v8f acc = {};
        #pragma unroll
        for (int kc = 0; kc < 32; ++kc) {
            v2f a = *(const v2f*)(arow + 4 * kc);
            v2f b = *(const v2f*)(brow + 4 * kc);
            acc = __builtin_amdgcn_wmma_f32_16x16x4_f32(false, a, false, b,
                                                        (short)0, acc, false, false);
        }
        // D layout: VGPR j, lanes 0-15 -> M=j, lanes 16-31 -> M=j+8; take column N=0
        if ((lane & 15) == 0) {
            int mb = (lane >> 4) * 8;
            #pragma unroll
            for (int j = 0; j < 8; ++j) lpart[kq * ROWS + 16 * g + mb + j] = acc[j];
        }
        __syncthreads();

        // sum 4 K-quarter partials, mask tail rows
        if (tid < ROWS) {
            float lv = 0.0f;
            #pragma unroll
            for (int q = 0; q < 4; ++q) lv += lpart[q * ROWS + tid];
            if (rowBase + tid >= N) lv = -INFINITY;
            prob[tid] = lv;
        }
        __syncthreads();

        // online softmax update (redundant per-thread scalars; LDS broadcast reads)
        float lmax = -INFINITY;
        for (int r = 0; r < ROWS; ++r) lmax = fmaxf(lmax, prob[r]);
        float m_new = fmaxf(m_run, lmax);
        float scale = (m_run == -INFINITY) ? 0.0f : expf(m_run - m_new);
        s_run *= scale; ctx0 *= scale; ctx1 *= scale;
        __syncthreads();
        if (tid < ROWS) {
            float lv = prob[tid];
            prob[tid] = (lv == -INFINITY) ? 0.0f : expf(lv - m_new);
        }
        __syncthreads();

        // rank-1 accumulation: thread owns dims (d0, d0+1); conflict-free b64 LDS reads
        float sloc = 0.0f;
        const float* xsd = xs + d0 + dadj;
        for (int r = 0; r < ROWS; ++r) {
            float p = prob[r];
            sloc += p;
            v2f xv = *(const v2f*)(xsd + r * LSTRIDE);
            ctx0 += p * xv.x;
            ctx1 += p * xv.y;
        }
        s_run += sloc;
        m_run = m_new;
        __syncthreads();   // all reads of this buffer done before TDM refills it
    }

    if (tid == 0) { ws[PM_OFF + blockIdx.x] = m_run; ws[PS_OFF + blockIdx.x] = s_run; }
    ws[PCTX_OFF + blockIdx.x * DDIM + d0]     = ctx0;
    ws[PCTX_OFF + blockIdx.x * DDIM + d0 + 1] = ctx1;
}

// Merge the NB flash-style partials into the step's context vector.
__global__ void s2s_combine_kernel(float* ws, int step) {
    const int tid = threadIdx.x;           // 256
    const float* pm = ws + PM_OFF;
    const float* ps = ws + PS_OFF;
    const float* pc = ws + PCTX_OFF;
    float M = -INFINITY;
    for (int b = 0; b < NB; ++b) M = fmaxf(M, pm[b]);
    float S = 0.0f;
    for (int b = 0; b < NB; ++b)
        S += ((pm[b] == -INFINITY) ? 0.0f : expf(pm[b] - M)) * ps[b];
    float inv = 1.0f / S;
    int d0 = 2 * tid;
    float c0 = 0.0f, c1 = 0.0f;
    for (int b = 0; b < NB; ++b) {
        float w = (pm[b] == -INFINITY) ? 0.0f : expf(pm[b] - M);
        c0 += w * pc[b * DDIM + d0];
        c1 += w * pc[b * DDIM + d0 + 1];
    }
    float* ctx = ws + CTX_OFF + step * DDIM;
    ctx[d0]     = c0 * inv;
    ctx[d0 + 1] = c1 * inv;
}

// gates = W_ih @ [ctx; h] + b_ih + W_hh @ h + b_hh ; one wave per gate row.
__global__ void s2s_lstm_gates_kernel(const float* __restrict__ W_ih,
                                      const float* __restrict__ W_hh,
                                      const float* __restrict__ b_ih,
                                      const float* __restrict__ b_hh,
                                      float* ws, int step) {
    const int lane = threadIdx.x & 31;
    const int wave = threadIdx.x >> 5;
    const int g = blockIdx.x * 8 + wave;   // 256 blocks * 8 waves = 2048 rows
    const float* ctx = ws + CTX_OFF + step * DDIM;
    const float* h   = ws + H_OFF;
    const float* wih = W_ih + (long)g * (2 * DDIM);
    const float* whh = W_hh + (long)g * DDIM;
    float acc = 0.0f;
    for (int j = lane; j < DDIM; j += 32) acc += wih[j] * ctx[j];
    for (int j = lane; j < DDIM; j += 32) acc += wih[DDIM + j] * h[j];
    for (int j = lane; j < DDIM; j += 32) acc += whh[j] * h[j];
    for (int off = 16; off > 0; off >>= 1) acc += __shfl_down(acc, off, 32);
    if (lane == 0) ws[GATES_OFF + g] = acc + b_ih[g] + b_hh[g];
}

__global__ void s2s_lstm_update_kernel(float* ws) {
    int d = threadIdx.x;                   // 512
    const float* gates = ws + GATES_OFF;
    float ig = 1.0f / (1.0f + expf(-gates[d]));
    float fg = 1.0f / (1.0f + expf(-gates[DDIM + d]));
    float gg = tanhf(gates[2 * DDIM + d]);
    float og = 1.0f / (1.0f + expf(-gates[3 * DDIM + d]));
    float c = fg * ws[C_OFF + d] + ig * gg;
    ws[C_OFF + d] = c;
    ws[H_OFF + d] = og * tanhf(c);
}

__global__ void s2s_proj_logits_kernel(const float* __restrict__ W_proj,
                                       const float* __restrict__ b_proj,
                                       float* ws) {
    const int lane = threadIdx.x & 31;
    const int wave = threadIdx.x >> 5;
    const int g = blockIdx.x * 8 + wave;   // 64 blocks * 8 waves = 512 rows
    const float* ctxall = ws + CTX_OFF;    // [3072]
    const float* wrow = W_proj + (long)g * (DDIM * NSTEPS);
    float acc = 0.0f;
    for (int j = lane; j < DDIM * NSTEPS; j += 32) acc += wrow[j] * ctxall[j];
    for (int off = 16; off > 0; off >>= 1) acc += __shfl_down(acc, off, 32);
    if (lane == 0) ws[PLOG_OFF + g] = acc + b_proj[g];
}

__global__ void s2s_proj_softmax_kernel(const float* __restrict__ ws, float* __restrict__ out) {
    __shared__ float sl[DDIM];
    int d = threadIdx.x;                   // 512
    sl[d] = ws[PLOG_OFF + d];
    __syncthreads();
    float M = -INFINITY;
    for (int j = 0; j < DDIM; ++j) M = fmaxf(M, sl[j]);
    float S = 0.0f;
    for (int j = 0; j < DDIM; ++j) S += expf(sl[j] - M);
    out[d] = expf(sl[d] - M) / S;
}

extern "C" void kernel_launch(void* const* d_in, const int* in_sizes, int n_in,
                              void* d_out, int out_size, void* d_ws, size_t ws_size,
                              hipStream_t stream) {
    const float* x      = (const float*)d_in[0];
    const float* W_ih   = (const float*)d_in[1];
    const float* W_hh   = (const float*)d_in[2];
    const float* b_ih   = (const float*)d_in[3];
    const float* b_hh   = (const float*)d_in[4];
    const float* W_proj = (const float*)d_in[5];
    const float* b_proj = (const float*)d_in[6];
    float* ws  = (float*)d_ws;
    float* out = (float*)d_out;
    const int N = in_sizes[0] / DDIM;

    const size_t smem = (size_t)(2 * XBUF + DDIM + 4 * ROWS + ROWS) * sizeof(float);

    s2s_init_kernel<<<1, 1024, 0, stream>>>(ws);
    for (int s = 0; s < NSTEPS; ++s) {
        s2s_attend_kernel<<<NB, 256, smem, stream>>>(x, ws, N);
        s2s_combine_kernel<<<1, 256, 0, stream>>>(ws, s);
        s2s_lstm_gates_kernel<<<256, 256, 0, stream>>>(W_ih, W_hh, b_ih, b_hh, ws, s);
        s2s_lstm_update_kernel<<<1, DDIM, 0, stream>>>(ws);
    }
    s2s_proj_logits_kernel<<<64, 256, 0, stream>>>(W_proj, b_proj, ws);
    s2s_proj_softmax_kernel<<<1, DDIM, 0, stream>>>(ws, out);
}